// KLSimMat_1872605741364
// MI455X (gfx1250) — compile-verified
//
#include <hip/hip_runtime.h>

// ---------------------------------------------------------------------------
// KL-similarity matrices (pos | rev | neg), N=512, D=768, out = 512 x 1536 f32.
//
// 1+kl(i,j,d) = P_i*A_j - Q_i*B_j + R_i - C_j    (all per-(row,d) factors)
//   i-side: P = 0.5/s2^2, Q = m2/s2^2, R = 0.5*m2^2/s2^2 + ln(s2) + 0.5
//   j-side: A = m1^2+s1^2, B = m1,     C = ln(s1)
// out[i,j] = (1/D) * sum_d rcp(1+kl)
// Compute-bound: 2 FMA + 1 SUB + 1 V_RCP_F32 + 1 ADD per (i,j,d) element.
// Staging uses CDNA5 async global->LDS (ASYNCcnt) with double buffering.
// ---------------------------------------------------------------------------

constexpr int kN   = 512;
constexpr int kD   = 768;
constexpr int TI   = 64;      // output tile rows per block
constexpr int TJ   = 64;      // output tile cols per block
constexpr int DK   = 16;      // d-chunk staged in LDS
constexpr int NC   = kD / DK; // 48 chunks
constexpr int OUTW = 3 * kN;  // 1536 output columns

#if __has_builtin(__builtin_amdgcn_global_load_async_to_lds_b128) && \
    __has_builtin(__builtin_amdgcn_s_wait_asynccnt)
#define CDNA5_ASYNC 1
typedef int v4i __attribute__((__vector_size__(16)));
typedef __attribute__((address_space(1))) v4i* as1_v4i_t;
typedef __attribute__((address_space(3))) v4i* as3_v4i_t;

__device__ __forceinline__ void async_copy_b128(const float* g, float* l) {
  // global_load_async_to_lds_b128: per-lane 16B, tracked by ASYNCcnt
  __builtin_amdgcn_global_load_async_to_lds_b128((as1_v4i_t)g, (as3_v4i_t)l,
                                                 /*offset=*/0, /*cpol=*/0);
}
#endif

__global__ __launch_bounds__(256) void klsim_tiled(
    const float* __restrict__ pre_mu, const float* __restrict__ pre_std,
    const float* __restrict__ ent_mu, const float* __restrict__ ent_std,
    const float* __restrict__ con_mu, const float* __restrict__ con_std,
    float* __restrict__ out)
{
  // Which of the three matrices this z-slice computes.
  const int z = blockIdx.z;
  const float *mu1, *sd1, *mu2, *sd2;
  int colOff;
  if (z == 0)      { mu1 = ent_mu; sd1 = ent_std; mu2 = pre_mu; sd2 = pre_std; colOff = 0;      }
  else if (z == 1) { mu1 = pre_mu; sd1 = pre_std; mu2 = ent_mu; sd2 = ent_std; colOff = kN;     }
  else             { mu1 = con_mu; sd1 = con_std; mu2 = pre_mu; sd2 = pre_std; colOff = 2 * kN; }

  const int i0 = blockIdx.y * TI;
  const int j0 = blockIdx.x * TJ;

  // Factor arrays, [d][row] so inner-loop reads are contiguous float4.
  __shared__ __align__(16) float sP[DK][TI];
  __shared__ __align__(16) float sQ[DK][TI];
  __shared__ __align__(16) float sR[DK][TI];
  __shared__ __align__(16) float sA[DK][TJ];
  __shared__ __align__(16) float sB[DK][TJ];
  __shared__ __align__(16) float sC[DK][TJ];

#ifdef CDNA5_ASYNC
  // Double-buffered raw staging tiles: [buf][m2,s2,m1,s1][row][d]  (2 x 16 KB)
  __shared__ __align__(16) float raw[2][4][TI][DK];
#endif

  const int tid = threadIdx.x;     // 0..255
  const int tx4 = (tid & 15) * 4;  // 4 consecutive j's
  const int ty4 = (tid >> 4) * 4;  // 4 consecutive i's
  const int sr  = tid >> 2;        // staging row 0..63
  const int sq4 = (tid & 3) * 4;   // staging d-quad offset 0,4,8,12

  float acc[4][4];
#pragma unroll
  for (int a = 0; a < 4; ++a)
#pragma unroll
    for (int b = 0; b < 4; ++b) acc[a][b] = 0.0f;

#ifdef CDNA5_ASYNC
  auto issue_chunk = [&](int d0c, int b) {
    const size_t o2 = (size_t)(i0 + sr) * kD + d0c + sq4;
    const size_t o1 = (size_t)(j0 + sr) * kD + d0c + sq4;
    async_copy_b128(&mu2[o2], &raw[b][0][sr][sq4]);
    async_copy_b128(&sd2[o2], &raw[b][1][sr][sq4]);
    async_copy_b128(&mu1[o1], &raw[b][2][sr][sq4]);
    async_copy_b128(&sd1[o1], &raw[b][3][sr][sq4]);
  };
  issue_chunk(0, 0);  // prime the pipeline
#endif

  for (int c = 0; c < NC; ++c) {
    const int d0 = c * DK;
    float4 m2v, s2v, m1v, s1v;

#ifdef CDNA5_ASYNC
    // Kick off next chunk's DMA into the other buffer, then retire this
    // chunk's 4 async ops (they complete in order; <=4 outstanding left).
    if (c + 1 < NC) {
      issue_chunk((c + 1) * DK, (c + 1) & 1);
      __builtin_amdgcn_s_wait_asynccnt(4);
    } else {
      __builtin_amdgcn_s_wait_asynccnt(0);
    }
    __syncthreads();  // all waves' async portions landed; factor LDS free
    {
      const int b = c & 1;
      m2v = *(const float4*)&raw[b][0][sr][sq4];
      s2v = *(const float4*)&raw[b][1][sr][sq4];
      m1v = *(const float4*)&raw[b][2][sr][sq4];
      s1v = *(const float4*)&raw[b][3][sr][sq4];
    }
#else
    const int d0n = d0 + DK;
    if (d0n < kD) {
      __builtin_prefetch(&mu2[(size_t)(i0 + sr) * kD + d0n + sq4], 0, 0);
      __builtin_prefetch(&sd2[(size_t)(i0 + sr) * kD + d0n + sq4], 0, 0);
      __builtin_prefetch(&mu1[(size_t)(j0 + sr) * kD + d0n + sq4], 0, 0);
      __builtin_prefetch(&sd1[(size_t)(j0 + sr) * kD + d0n + sq4], 0, 0);
    }
    __syncthreads();  // previous inner loop done reading factor LDS
    m2v = *(const float4*)&mu2[(size_t)(i0 + sr) * kD + d0 + sq4];
    s2v = *(const float4*)&sd2[(size_t)(i0 + sr) * kD + d0 + sq4];
    m1v = *(const float4*)&mu1[(size_t)(j0 + sr) * kD + d0 + sq4];
    s1v = *(const float4*)&sd1[(size_t)(j0 + sr) * kD + d0 + sq4];
#endif

    // ---- compute factors -> LDS -----------------------------------------
    {
      const float mv[4] = {m2v.x, m2v.y, m2v.z, m2v.w};
      const float sv[4] = {s2v.x, s2v.y, s2v.z, s2v.w};
#pragma unroll
      for (int k = 0; k < 4; ++k) {
        const float r2 = __builtin_amdgcn_rcpf(sv[k] * sv[k]);  // 1/s2^2
        const float P  = 0.5f * r2;
        const float Q  = r2 * mv[k];
        const float R  = fmaf(P, mv[k] * mv[k], __logf(sv[k]) + 0.5f);
        const int d = sq4 + k;
        sP[d][sr] = P; sQ[d][sr] = Q; sR[d][sr] = R;
      }
    }
    {
      const float mv[4] = {m1v.x, m1v.y, m1v.z, m1v.w};
      const float sv[4] = {s1v.x, s1v.y, s1v.z, s1v.w};
#pragma unroll
      for (int k = 0; k < 4; ++k) {
        const float A = fmaf(mv[k], mv[k], sv[k] * sv[k]);
        const float B = mv[k];
        const float C = __logf(sv[k]);
        const int d = sq4 + k;
        sA[d][sr] = A; sB[d][sr] = B; sC[d][sr] = C;
      }
    }
    __syncthreads();

    // ---- inner loop: 16 d-values x 4x4 register tile --------------------
#pragma unroll
    for (int d = 0; d < DK; ++d) {
      const float4 Pv = *(const float4*)&sP[d][ty4];
      const float4 Qv = *(const float4*)&sQ[d][ty4];
      const float4 Rv = *(const float4*)&sR[d][ty4];
      const float4 Av = *(const float4*)&sA[d][tx4];
      const float4 Bv = *(const float4*)&sB[d][tx4];
      const float4 Cv = *(const float4*)&sC[d][tx4];
      const float Pa[4] = {Pv.x, Pv.y, Pv.z, Pv.w};
      const float Qa[4] = {Qv.x, Qv.y, Qv.z, Qv.w};
      const float Ra[4] = {Rv.x, Rv.y, Rv.z, Rv.w};
      const float Aa[4] = {Av.x, Av.y, Av.z, Av.w};
      const float Ba[4] = {Bv.x, Bv.y, Bv.z, Bv.w};
      const float Ca[4] = {Cv.x, Cv.y, Cv.z, Cv.w};
#pragma unroll
      for (int ii = 0; ii < 4; ++ii) {
#pragma unroll
        for (int jj = 0; jj < 4; ++jj) {
          const float den = fmaf(Pa[ii], Aa[jj], Ra[ii]) - fmaf(Qa[ii], Ba[jj], Ca[jj]);
          acc[ii][jj] += __builtin_amdgcn_rcpf(den);  // v_rcp_f32, trans pipe
        }
      }
    }
  }

  // ---- write 4 rows x float4 --------------------------------------------
  const float inv = 1.0f / (float)kD;
#pragma unroll
  for (int ii = 0; ii < 4; ++ii) {
    float4 o;
    o.x = acc[ii][0] * inv; o.y = acc[ii][1] * inv;
    o.z = acc[ii][2] * inv; o.w = acc[ii][3] * inv;
    *(float4*)&out[(size_t)(i0 + ty4 + ii) * OUTW + colOff + j0 + tx4] = o;
  }
}

// ---------------------------------------------------------------------------
extern "C" void kernel_launch(void* const* d_in, const int* in_sizes, int n_in,
                              void* d_out, int out_size, void* d_ws, size_t ws_size,
                              hipStream_t stream) {
  const float* pre_mu  = (const float*)d_in[0];
  const float* pre_std = (const float*)d_in[1];
  const float* ent_mu  = (const float*)d_in[2];
  const float* ent_std = (const float*)d_in[3];
  const float* con_mu  = (const float*)d_in[4];
  const float* con_std = (const float*)d_in[5];
  float* out = (float*)d_out;

  dim3 grid(kN / TJ, kN / TI, 3);  // (8, 8, 3)
  klsim_tiled<<<grid, 256, 0, stream>>>(pre_mu, pre_std, ent_mu, ent_std,
                                        con_mu, con_std, out);
}